// NeuralDisCoCirc_86294482911365
// MI455X (gfx1250) — compile-verified
//
#include <hip/hip_runtime.h>

typedef __attribute__((ext_vector_type(2))) float v2f;
typedef __attribute__((ext_vector_type(4))) float v4f;
typedef __attribute__((ext_vector_type(8))) float v8f;

#define DIM   1024
#define BATCH 32
#define NLAYERS 8

// One layer of the scan: hout[b,j] = maskedrelu( sum_i hin[b,i]*W[b,i,j] + bias[b,j] )
// grid = (8 col-blocks of 128, 32 examples), block = 256 threads (8 waves).
// Each wave computes 16 output columns via V_WMMA_F32_16X16X4_F32 with the
// TRANSPOSED mapping: A = W^T tile (16 cols x 4 k), B = h[k..k+3] broadcast
// into all 16 N columns. D[m, n] = out[col0+m] for every n, so operand
// construction needs no lane masking at all.
__global__ __launch_bounds__(256) void dense_layer_wmma_f32(
    const float* __restrict__ hin,   // [B, D]
    const float* __restrict__ W,     // [B, D, D]  (row i, col j contiguous in j)
    const float* __restrict__ bias,  // [B, D]
    const int*   __restrict__ mask,  // [B, D]
    float* __restrict__ hout)        // [B, D]
{
    __shared__ float hs[DIM];

    const int b   = blockIdx.y;
    const int tid = threadIdx.x;

    // Cooperative load of h[b,:] into LDS: 256 threads x float4 = 1024 floats.
    {
        const v4f* src = (const v4f*)(hin + (size_t)b * DIM);
        ((v4f*)hs)[tid] = src[tid];
    }
    __syncthreads();

    const int wave = tid >> 5;
    const int lane = tid & 31;
    const int col0 = blockIdx.x * 128 + wave * 16;   // 16 output columns per wave
    const int half = lane >> 4;                      // 0: K rows k,k+1   1: K rows k+2,k+3
    const int m    = lane & 15;                      // output column (M row of A)

    // Per-lane weight column base: W[b, :, col0+m]
    const float* Wb = W + (size_t)b * DIM * DIM + (size_t)col0 + m;

    v8f c = {};  // D[m, n] replicated over n; rows M=0..15 = out[col0 .. col0+15]

    for (int kk = 0; kk < DIM; kk += 64) {
        // Prefetch the next 64-row weight chunk (speculative; dropped if OOB).
        __builtin_prefetch(Wb + (size_t)(kk + 64 + 2 * half) * DIM, 0, 3);

#pragma unroll
        for (int k2 = 0; k2 < 64; k2 += 4) {
            const int k = kk + k2;

            // A tile 16x4 f32 (documented layout): lane m, VGPR0 = K=(0|2),
            // VGPR1 = K=(1|3).  A[m,kk] = W[k+kk, col0+m].
            const float* p = Wb + (size_t)(k + 2 * half) * DIM;
            v2f a;
            a.x = p[0];     // W[k   + 2*half, col0+m]
            a.y = p[DIM];   // W[k+1 + 2*half, col0+m]

            // B tile 4x16 f32: B[kk, n] = h[k+kk] for every n — pure broadcast,
            // one conflict-free ds_load_b64 per half-wave, no masking.
            const v2f bh = *(const v2f*)(hs + k + 2 * half);

            // D = A x B + C  (v_wmma_f32_16x16x4_f32)
            c = __builtin_amdgcn_wmma_f32_16x16x4_f32(
                    /*neg_a=*/false, a, /*neg_b=*/false, bh,
                    /*c_mod=*/(short)0, c, /*reuse_a=*/false, /*reuse_b=*/false);
        }
    }

    // C/D layout: VGPR r, lanes 0-15 hold (M=r, N=lane); lanes 16-31 hold
    // (M=8+r, N=lane-16).  Since every N column is identical, lanes 0-15 all
    // hold out[col0+r] in c[r], lanes 16-31 all hold out[col0+8+r] in c[r].
    // Lanes 0-7 store columns col0+0..7; lanes 16-23 store columns col0+8..15.
    {
        const int r = lane & 7;
        float v = c[0];
#pragma unroll
        for (int i = 1; i < 8; ++i)
            if (r == i) v = c[i];

        if ((lane & 15) < 8) {
            const int j   = col0 + (half << 3) + r;
            const int idx = b * DIM + j;
            float o = v + bias[idx];
            if (mask[idx]) o = fmaxf(o, 0.0f);
            hout[idx] = o;
        }
    }
}

extern "C" void kernel_launch(void* const* d_in, const int* in_sizes, int n_in,
                              void* d_out, int out_size, void* d_ws, size_t ws_size,
                              hipStream_t stream) {
    const float* x    = (const float*)d_in[0];  // [B, D]
    const float* W    = (const float*)d_in[1];  // [L, B, D, D]
    const float* bias = (const float*)d_in[2];  // [L, B, D]
    const int*   mask = (const int*)d_in[3];    // [L, B, D]
    float*       out  = (float*)d_out;          // [B, D]
    float*       hbuf = (float*)d_ws;           // ping-pong: 2 * B*D floats

    const size_t layerW = (size_t)BATCH * DIM * DIM;
    const size_t layerV = (size_t)BATCH * DIM;

    dim3 grid(DIM / 128, BATCH);
    dim3 block(256);

    const float* cur = x;
    for (int l = 0; l < NLAYERS; ++l) {
        float* o = (l == NLAYERS - 1) ? out : hbuf + (size_t)(l & 1) * layerV;
        hipLaunchKernelGGL(dense_layer_wmma_f32, grid, block, 0, stream,
                           cur,
                           W + (size_t)l * layerW,
                           bias + (size_t)l * layerV,
                           mask + (size_t)l * layerV,
                           o);
        cur = o;
    }
}